// CapsuleLayerTSV_82497731822168
// MI455X (gfx1250) — compile-verified
//
#include <hip/hip_runtime.h>
#include <hip/hip_bf16.h>
#include <math.h>

typedef float v2f __attribute__((ext_vector_type(2)));
typedef float v8f __attribute__((ext_vector_type(8)));

#define ADAPTERS 40
#define CAPS     3
#define CLASS_DIM 200
#define IN_CH    600
#define BATCH    256
#define DBIG     768
#define NTILES_N 13       // ceil(200/16)
#define NPAD     208      // 13*16, padded N width in LDS
#define KC       24       // K-chunk staged in LDS (600 = 25 * 24)

#if defined(__has_builtin)
#  if __has_builtin(__builtin_amdgcn_global_load_async_to_lds_b32)
#    define HAVE_ASYNC_LDS 1
#  endif
#endif
#ifndef HAVE_ASYNC_LDS
#  define HAVE_ASYNC_LDS 0
#endif

typedef __attribute__((address_space(1))) int* gptr_i32;
typedef __attribute__((address_space(3))) int* lptr_i32;

__device__ __forceinline__ v8f wmma_f32(v2f a, v2f b, v8f c) {
    return __builtin_amdgcn_wmma_f32_16x16x4_f32(
        false, a, false, b, (short)0, c, false, false);
}

// ---------------------------------------------------------------------------
// Kernel A: priors[n][b][k][d] = sum_c x[b][k][c] * W[k][n][c][d]
// One block (256 thr, 8 waves) per (kA, n, mGroup): block covers 128 M-rows
// and the FULL N=200 (13 tiles). Each wave keeps 13 accumulators in VGPRs so
// every A fragment (loaded once from global) feeds 13 WMMAs. B K-chunks
// (24 x 208 floats, ~20 KB) staged in LDS via async global->LDS loads and
// shared by all 8 waves. Padded N columns (d>=200) use clamped in-bounds
// addresses; their accumulator columns are never stored.
// ---------------------------------------------------------------------------
__global__ __launch_bounds__(256) void priors_gemm_kernel(
    const float* __restrict__ x,        // [256][40][600]
    const float* __restrict__ W,        // [40][3][600][200]
    float* __restrict__ priors)         // [3][256][40][200]
{
    __shared__ float Blds[KC * NPAD];   // 19968 B

    const int tid  = threadIdx.x;
    const int lane = tid & 31;
    const int wave = tid >> 5;
    const int half = lane >> 4;
    const int l    = lane & 15;

    const int mg = blockIdx.x & 1;            // M group: rows [mg*128, mg*128+128)
    const int kn = blockIdx.x >> 1;           // 0..119
    const int kA = kn % ADAPTERS;
    const int n  = kn / ADAPTERS;

    const int m0 = mg * 128 + wave * 16;      // this wave's 16 rows

    const float* __restrict__ Arow = x + (size_t)(m0 + l) * (ADAPTERS * IN_CH)
                                       + (size_t)kA * IN_CH;
    const float* __restrict__ Bg   = W + ((size_t)(kA * CAPS + n)) * IN_CH * CLASS_DIM;

    v8f acc[NTILES_N];
    #pragma unroll
    for (int nt = 0; nt < NTILES_N; ++nt) acc[nt] = (v8f){};

    for (int c0 = 0; c0 < IN_CH; c0 += KC) {
        // ---- stage B chunk [c0 .. c0+KC) x [0 .. 208) into LDS ----
        for (int idx = tid; idx < KC * NPAD; idx += 256) {
            const int kk = idx / NPAD;
            const int dl = idx - kk * NPAD;
            const int dc = (dl < CLASS_DIM) ? dl : (CLASS_DIM - 1);
            const size_t goff = (size_t)(c0 + kk) * CLASS_DIM + dc;
#if HAVE_ASYNC_LDS
            __builtin_amdgcn_global_load_async_to_lds_b32(
                (gptr_i32)(Bg + goff),
                (lptr_i32)&Blds[idx], 0, 0);
#else
            Blds[idx] = Bg[goff];
#endif
        }
#if HAVE_ASYNC_LDS
        asm volatile("s_wait_asynccnt 0" ::: "memory");
#endif
        __syncthreads();

        // ---- consume: 6 K-steps; 1 A-frag load feeds 13 WMMAs ----
        #pragma unroll
        for (int s = 0; s < KC / 4; ++s) {
            const int ks = 4 * s;
            const int ca = c0 + ks + 2 * half;
            v2f a;
            a.x = Arow[ca];
            a.y = Arow[ca + 1];
            const int rb0 = (ks + 2 * half) * NPAD + l;
            #pragma unroll
            for (int nt = 0; nt < NTILES_N; ++nt) {
                v2f b;
                b.x = Blds[rb0 + nt * 16];
                b.y = Blds[rb0 + NPAD + nt * 16];
                acc[nt] = wmma_f32(a, b, acc[nt]);
            }
        }
        __syncthreads();
    }

    // ---- store: 13 tiles x 8 rows per lane, guard d < 200 ----
    const size_t rowStride = (size_t)ADAPTERS * CLASS_DIM;        // next b
    float* outBase = priors + (((size_t)n * BATCH + (m0 + 8 * half)) * ADAPTERS + kA)
                              * CLASS_DIM;
    #pragma unroll
    for (int nt = 0; nt < NTILES_N; ++nt) {
        const int d = nt * 16 + l;
        if (d < CLASS_DIM) {
            #pragma unroll
            for (int r = 0; r < 8; ++r)
                outBase[(size_t)r * rowStride + d] = acc[nt][r];
        }
    }
}

// ---------------------------------------------------------------------------
// Block-wide sum over 256 threads (wave32). red must have >= 9 floats.
// ---------------------------------------------------------------------------
__device__ __forceinline__ float blockSum256(float v, float* red)
{
    #pragma unroll
    for (int o = 16; o > 0; o >>= 1) v += __shfl_down(v, o, 32);
    const int lane = threadIdx.x & 31;
    const int w    = threadIdx.x >> 5;
    if (lane == 0) red[w] = v;
    __syncthreads();
    if (threadIdx.x == 0) {
        float s = 0.0f;
        #pragma unroll
        for (int i = 0; i < 8; ++i) s += red[i];
        red[8] = s;
    }
    __syncthreads();
    float r = red[8];
    __syncthreads();
    return r;
}

// ---------------------------------------------------------------------------
// Kernel B: 3 routing iterations per (n,b). One 256-thread block per (n,b).
// Writes final (raw, pre-squash) vote[n][b][d] to workspace.
// ---------------------------------------------------------------------------
__global__ __launch_bounds__(256) void routing_kernel(
    const float* __restrict__ priors,   // [3][256][40][200]
    const float* __restrict__ tsv,      // [40][40]
    const int*   __restrict__ tptr,
    float* __restrict__ voteOut)        // [3][256][200]
{
    const int nb  = blockIdx.x;                 // n*256 + b
    const int tid = threadIdx.x;
    const int lane = tid & 31;
    const int w    = tid >> 5;
    const int t = *tptr;

    __shared__ float logits[ADAPTERS];
    __shared__ float p[ADAPTERS];
    __shared__ float tsvv[ADAPTERS];
    __shared__ float maskv[ADAPTERS];
    __shared__ float outv[CLASS_DIM];
    __shared__ float red[9];

    const float* __restrict__ P = priors + (size_t)nb * ADAPTERS * CLASS_DIM;

    if (tid < ADAPTERS) {
        float tv = tsv[t * ADAPTERS + tid];
        tsvv[tid]  = tv;
        maskv[tid] = (tv == 0.0f) ? -10000.0f : 0.0f;
        logits[tid] = 0.0f;
    }
    __syncthreads();

    float v = 0.0f;
    for (int it = 0; it < 3; ++it) {
        // masked softmax over k (logits constant along d, so scalar per k)
        if (tid == 0) {
            float m = -INFINITY;
            for (int k = 0; k < ADAPTERS; ++k) {
                float l2 = logits[k] * tsvv[k] + maskv[k];
                p[k] = l2;
                m = fmaxf(m, l2);
            }
            float s = 0.0f;
            for (int k = 0; k < ADAPTERS; ++k) {
                float e = __expf(p[k] - m);
                p[k] = e;
                s += e;
            }
            float inv = 1.0f / s;
            for (int k = 0; k < ADAPTERS; ++k) p[k] *= inv;
        }
        __syncthreads();

        // vote[d] = sum_k p[k] * priors[k][d]
        v = 0.0f;
        if (tid < CLASS_DIM) {
            for (int k = 0; k < ADAPTERS; ++k)
                v += p[k] * P[(size_t)k * CLASS_DIM + tid];
        }
        if (it == 2) break;

        // squash: out = v * sqrt(sq) / (1 + sq)
        float sq = blockSum256(v * v, red);
        float f  = sqrtf(sq) / (1.0f + sq);
        if (tid < CLASS_DIM) outv[tid] = v * f;
        __syncthreads();

        // logits[k] += dot(priors[k,:], out) ; 8 waves x 5 k's each
        for (int kk = 0; kk < 5; ++kk) {
            int k = w * 5 + kk;
            float a = 0.0f;
            for (int dd = lane; dd < CLASS_DIM; dd += 32)
                a += P[(size_t)k * CLASS_DIM + dd] * outv[dd];
            #pragma unroll
            for (int o = 16; o > 0; o >>= 1) a += __shfl_down(a, o, 32);
            if (lane == 0) logits[k] += a;
        }
        __syncthreads();
    }

    if (tid < CLASS_DIM)
        voteOut[(size_t)nb * CLASS_DIM + tid] = v;
}

// ---------------------------------------------------------------------------
// Kernel C: out[b2,d2,dd] = (sum_n h_n * larger_w[dd,n] + larger_b[dd]) * g[dd]
// where h_n = voteflat[3*(b2*200+d2) + n]  (the reference's raw reshape),
// g[dd] = sigmoid(s * elarger[t][dd]). One block per (b2,d2), 768 threads.
// ---------------------------------------------------------------------------
__global__ __launch_bounds__(768) void expand_kernel(
    const float* __restrict__ vote,      // flat [3*256*200]
    const float* __restrict__ larger_w,  // [768][3]
    const float* __restrict__ larger_b,  // [768]
    const float* __restrict__ elarger,   // [40][768]
    const float* __restrict__ sptr,      // [1]
    const int*   __restrict__ tptr,
    float* __restrict__ out)             // [256][200][768]
{
    const int bd = blockIdx.x;           // b2*200 + d2
    const int dd = threadIdx.x;
    const int t = *tptr;

    __shared__ float h[3];
    if (threadIdx.x < 3) h[threadIdx.x] = vote[(size_t)bd * 3 + threadIdx.x];
    __syncthreads();

    float acc = h[0] * larger_w[dd * 3 + 0]
              + h[1] * larger_w[dd * 3 + 1]
              + h[2] * larger_w[dd * 3 + 2]
              + larger_b[dd];
    float g = 1.0f / (1.0f + __expf(-sptr[0] * elarger[t * DBIG + dd]));
    out[(size_t)bd * DBIG + dd] = acc * g;
}

// ---------------------------------------------------------------------------
extern "C" void kernel_launch(void* const* d_in, const int* in_sizes, int n_in,
                              void* d_out, int out_size, void* d_ws, size_t ws_size,
                              hipStream_t stream)
{
    const int*   t_p       = (const int*)  d_in[0];
    const float* x         = (const float*)d_in[1];
    const float* s_p       = (const float*)d_in[2];
    const float* route_w   = (const float*)d_in[3];
    const float* larger_w  = (const float*)d_in[4];
    const float* larger_b  = (const float*)d_in[5];
    const float* elarger   = (const float*)d_in[6];
    const float* tsv       = (const float*)d_in[7];
    float* out = (float*)d_out;

    float* priorsWS = (float*)d_ws;                                  // 3*256*40*200 floats
    float* voteWS   = priorsWS + (size_t)CAPS * BATCH * ADAPTERS * CLASS_DIM;

    // A: one block per (kA, n, mGroup) = 120 * 2 = 240 blocks
    priors_gemm_kernel<<<CAPS * ADAPTERS * 2, 256, 0, stream>>>(
        x, route_w, priorsWS);

    // B: one block per (n,b)
    routing_kernel<<<CAPS * BATCH, 256, 0, stream>>>(priorsWS, tsv, t_p, voteWS);

    // C: one block per (b2,d2)
    expand_kernel<<<BATCH * CLASS_DIM, DBIG, 0, stream>>>(
        voteWS, larger_w, larger_b, elarger, s_p, t_p, out);
}